// GCNBlock_62457414418469
// MI455X (gfx1250) — compile-verified
//
#include <hip/hip_runtime.h>

#define NN 50000
#define NE 600000
#define DIM 128
#define BN_EPS 1e-5f

typedef __attribute__((ext_vector_type(2))) float v2f;
typedef __attribute__((ext_vector_type(8))) float v8f;

// ---------------- degree / normalization ----------------

__global__ void init_deg_kernel(float* __restrict__ deg, int n) {
    int i = blockIdx.x * blockDim.x + threadIdx.x;
    if (i < n) deg[i] = 1.0f;  // implicit self-loop
}

__global__ void deg_count_kernel(const int* __restrict__ col, float* __restrict__ deg, int e) {
    int i = blockIdx.x * blockDim.x + threadIdx.x;
    if (i < e) unsafeAtomicAdd(&deg[col[i]], 1.0f);
}

__global__ void rsqrt_kernel(float* __restrict__ deg, int n) {
    int i = blockIdx.x * blockDim.x + threadIdx.x;
    if (i < n) deg[i] = rsqrtf(deg[i]);
}

__global__ void zero_kernel(float* __restrict__ p, int n) {
    int i = blockIdx.x * blockDim.x + threadIdx.x;
    if (i < n) p[i] = 0.0f;
}

// ---------------- 128x128 transpose (once per layer; W is tiny) ----------------

__global__ __launch_bounds__(256)
void transpose128_kernel(const float* __restrict__ W, float* __restrict__ Wt) {
    int t = blockIdx.x * blockDim.x + threadIdx.x;  // 0 .. 128*128-1
    int r = t >> 7;
    int c = t & 127;
    Wt[(size_t)c * DIM + r] = W[(size_t)r * DIM + c];
}

// ---------------- dense GEMM via f32 WMMA ----------------
// H[nrows x 128] = X[nrows x 128] @ W[128 x 128], with W supplied TRANSPOSED
// (Wt[c][k] = W[k][c]) so both A and B fragments are contiguous b64 loads.
// One wave computes one 16x16 output tile; 8 waves/block cover all 128 cols.
__global__ __launch_bounds__(256)
void gemm_wmma_kernel(const float* __restrict__ X, const float* __restrict__ Wt,
                      float* __restrict__ H) {
    const int tileM = blockIdx.x * 16;
    const int wave  = threadIdx.x >> 5;
    const int lane  = threadIdx.x & 31;
    const int tileN = wave * 16;
    const int mrow  = lane & 15;
    const int hi    = lane >> 4;  // 0: K={0,1}, 1: K={2,3} per 16x4 A layout

    v8f acc = {};
    const float* xr = X + (size_t)(tileM + mrow) * DIM + 2 * hi;   // lane's A row
    const float* wc = Wt + (size_t)(tileN + mrow) * DIM + 2 * hi;  // lane's B column

#pragma unroll
    for (int k0 = 0; k0 < DIM; k0 += 4) {
        v2f a = *(const v2f*)(xr + k0);   // A 16x4: VGPR0=K=ka, VGPR1=K=ka+1
        v2f b = *(const v2f*)(wc + k0);   // B 4x16: VGPR0=K=ka, VGPR1=K=ka+1
        acc = __builtin_amdgcn_wmma_f32_16x16x4_f32(
            /*neg_a=*/false, a, /*neg_b=*/false, b,
            /*c_mod=*/(short)0, acc, /*reuse_a=*/false, /*reuse_b=*/false);
    }

    // C/D 16x16 f32 layout: VGPR r -> M = r + 8*hi, N = tileN + mrow
    float* out = H + (size_t)(tileM + 8 * hi) * DIM + tileN + mrow;
#pragma unroll
    for (int r = 0; r < 8; ++r)
        out[(size_t)r * DIM] = acc[r];
}

// ---------------- self-loop init: agg = h * dis^2 ----------------

__global__ __launch_bounds__(256)
void selfloop_kernel(const float* __restrict__ h, const float* __restrict__ dis,
                     float* __restrict__ agg, int n) {
    int t = blockIdx.x * blockDim.x + threadIdx.x;  // over n*32 float4 slots
    if (t >= n * 32) return;
    int node = t >> 5;
    float s = dis[node];
    s = s * s;
    float4 v = ((const float4*)h)[t];
    v.x *= s; v.y *= s; v.z *= s; v.w *= s;
    ((float4*)agg)[t] = v;
}

// ---------------- edge scatter: agg[col] += h[row] * dis[row]*dis[col] ----------------
// One edge per wave; lane handles 4 consecutive features (fully coalesced).
__global__ __launch_bounds__(256)
void edge_scatter_kernel(const float* __restrict__ h, float* __restrict__ agg,
                         const int* __restrict__ row, const int* __restrict__ col,
                         const float* __restrict__ dis, int e) {
    int warp = threadIdx.x >> 5;
    int lane = threadIdx.x & 31;
    int eid = blockIdx.x * 8 + warp;
    if (eid >= e) return;
    int r = row[eid];
    int c = col[eid];
    float norm = dis[r] * dis[c];
    float4 v = ((const float4*)(h + (size_t)r * DIM))[lane];
    float* dst = agg + (size_t)c * DIM + lane * 4;
    unsafeAtomicAdd(dst + 0, v.x * norm);
    unsafeAtomicAdd(dst + 1, v.y * norm);
    unsafeAtomicAdd(dst + 2, v.z * norm);
    unsafeAtomicAdd(dst + 3, v.w * norm);
}

// ---------------- bias + relu ----------------

__global__ __launch_bounds__(256)
void bias_relu_kernel(const float* __restrict__ agg, const float* __restrict__ b,
                      float* __restrict__ out, int n) {
    int t = blockIdx.x * blockDim.x + threadIdx.x;
    if (t >= n * 32) return;
    int j = t & 31;
    float4 v = ((const float4*)agg)[t];
    float4 bb = ((const float4*)b)[j];
    v.x = fmaxf(v.x + bb.x, 0.0f);
    v.y = fmaxf(v.y + bb.y, 0.0f);
    v.z = fmaxf(v.z + bb.z, 0.0f);
    v.w = fmaxf(v.w + bb.w, 0.0f);
    ((float4*)out)[t] = v;
}

// ---------------- batchnorm ----------------

__global__ __launch_bounds__(128)
void bn_reduce_kernel(const float* __restrict__ h, float* __restrict__ sum,
                      float* __restrict__ sumsq, int n, int rows_per_block) {
    int d = threadIdx.x;  // feature
    int r0 = blockIdx.x * rows_per_block;
    int r1 = r0 + rows_per_block;
    if (r1 > n) r1 = n;
    float s = 0.0f, ss = 0.0f;
    for (int r = r0; r < r1; ++r) {
        float v = h[(size_t)r * DIM + d];
        s += v;
        ss += v * v;
    }
    unsafeAtomicAdd(&sum[d], s);
    unsafeAtomicAdd(&sumsq[d], ss);
}

__global__ __launch_bounds__(128)
void bn_finalize_kernel(const float* __restrict__ sum, const float* __restrict__ sumsq,
                        float* __restrict__ mean, float* __restrict__ istd, int n) {
    int d = threadIdx.x;
    float invn = 1.0f / (float)n;
    float m = sum[d] * invn;
    float var = sumsq[d] * invn - m * m;
    mean[d] = m;
    istd[d] = rsqrtf(var + BN_EPS);
}

__global__ __launch_bounds__(256)
void bn_norm_kernel(const float* __restrict__ h, const float* __restrict__ mean,
                    const float* __restrict__ istd, const float* __restrict__ gamma,
                    const float* __restrict__ beta, float* __restrict__ out, int n) {
    int t = blockIdx.x * blockDim.x + threadIdx.x;
    if (t >= n * 32) return;
    int j = t & 31;
    float4 v  = ((const float4*)h)[t];
    float4 m  = ((const float4*)mean)[j];
    float4 is = ((const float4*)istd)[j];
    float4 g  = ((const float4*)gamma)[j];
    float4 be = ((const float4*)beta)[j];
    float4 o;
    o.x = g.x * (v.x - m.x) * is.x + be.x;
    o.y = g.y * (v.y - m.y) * is.y + be.y;
    o.z = g.z * (v.z - m.z) * is.z + be.z;
    o.w = g.w * (v.w - m.w) * is.w + be.w;
    ((float4*)out)[t] = o;
}

// ---------------- launcher ----------------

extern "C" void kernel_launch(void* const* d_in, const int* in_sizes, int n_in,
                              void* d_out, int out_size, void* d_ws, size_t ws_size,
                              hipStream_t stream) {
    const float* x     = (const float*)d_in[0];
    const int*   eidx  = (const int*)d_in[1];   // (2, E): first E = row, next E = col
    const float* W1    = (const float*)d_in[2];
    const float* b1    = (const float*)d_in[3];
    const float* W2    = (const float*)d_in[4];
    const float* b2    = (const float*)d_in[5];
    const float* gamma = (const float*)d_in[6];
    const float* beta  = (const float*)d_in[7];
    float* out = (float*)d_out;

    const int* row = eidx;
    const int* col = eidx + NE;

    // Workspace layout (floats):
    //   bufA[N*D] | bufB[N*D] | dis[N] | sum[D] | sumsq[D] | mean[D] | istd[D] | W1t[D*D] | W2t[D*D]
    float* bufA  = (float*)d_ws;
    float* bufB  = bufA + (size_t)NN * DIM;
    float* dis   = bufB + (size_t)NN * DIM;
    float* sum   = dis + NN;
    float* sumsq = sum + DIM;
    float* mean  = sumsq + DIM;
    float* istd  = mean + DIM;
    float* W1t   = istd + DIM;
    float* W2t   = W1t + (size_t)DIM * DIM;

    const int T = 256;
    const int elem_blocks = (NN * 32 + T - 1) / T;      // float4-granular elementwise
    const int gemm_blocks = NN / 16;                    // 50000 % 16 == 0
    const int edge_blocks = (NE + 7) / 8;               // 8 edges per block (1/wave)
    const int tr_blocks   = (DIM * DIM) / T;            // 64
    const int bn_blocks = 256;
    const int rows_per_block = (NN + bn_blocks - 1) / bn_blocks;

    // --- degree / dis + weight transposes + accumulator zeroing ---
    init_deg_kernel<<<(NN + T - 1) / T, T, 0, stream>>>(dis, NN);
    deg_count_kernel<<<(NE + T - 1) / T, T, 0, stream>>>(col, dis, NE);
    rsqrt_kernel<<<(NN + T - 1) / T, T, 0, stream>>>(dis, NN);
    zero_kernel<<<1, 2 * DIM, 0, stream>>>(sum, 2 * DIM);  // sum + sumsq contiguous
    transpose128_kernel<<<tr_blocks, T, 0, stream>>>(W1, W1t);
    transpose128_kernel<<<tr_blocks, T, 0, stream>>>(W2, W2t);

    // --- layer 1 ---
    gemm_wmma_kernel<<<gemm_blocks, T, 0, stream>>>(x, W1t, bufA);           // h1 = x@W1
    selfloop_kernel<<<elem_blocks, T, 0, stream>>>(bufA, dis, bufB, NN);     // agg = h1*dis^2
    edge_scatter_kernel<<<edge_blocks, T, 0, stream>>>(bufA, bufB, row, col, dis, NE);
    bias_relu_kernel<<<elem_blocks, T, 0, stream>>>(bufB, b1, bufA, NN);     // h1' = relu(agg+b1)

    // --- layer 2 ---
    gemm_wmma_kernel<<<gemm_blocks, T, 0, stream>>>(bufA, W2t, bufB);        // h2 = h1'@W2
    selfloop_kernel<<<elem_blocks, T, 0, stream>>>(bufB, dis, bufA, NN);     // agg2 = h2*dis^2
    edge_scatter_kernel<<<edge_blocks, T, 0, stream>>>(bufB, bufA, row, col, dis, NE);
    bias_relu_kernel<<<elem_blocks, T, 0, stream>>>(bufA, b2, bufB, NN);     // h2' = relu(agg2+b2)

    // --- batchnorm ---
    bn_reduce_kernel<<<bn_blocks, DIM, 0, stream>>>(bufB, sum, sumsq, NN, rows_per_block);
    bn_finalize_kernel<<<1, DIM, 0, stream>>>(sum, sumsq, mean, istd, NN);
    bn_norm_kernel<<<elem_blocks, T, 0, stream>>>(bufB, mean, istd, gamma, beta, out, NN);
}